// GraphConvolutionalLSTM_base0_86285892976678
// MI455X (gfx1250) — compile-verified
//
#include <hip/hip_runtime.h>
#include <hip/hip_bf16.h>
#include <cmath>

// Problem dims (compile-time, match reference)
#define ROI   200
#define TT    100
#define HH    100
#define BB    64
#define G4    400          // 4*H
#define NROWS 6400         // B*T
#define EPSV  1e-5f

typedef __attribute__((ext_vector_type(2))) float v2f;
typedef __attribute__((ext_vector_type(4))) float v4f;
typedef __attribute__((ext_vector_type(8))) float v8f;

__device__ __forceinline__ float sigmoidf_(float x) { return 1.0f / (1.0f + __expf(-x)); }

// ---------------------------------------------------------------------------
// K1: GCN  feat[b,t,i] = relu( sum_j x[b,t,i,j] * w[j] + b0 )
// One block per (b,t) tile; wave-per-row, lane-strided b128 nontemporal loads
// (x is 1.024 GB streamed exactly once -> don't pollute L2, minimize VMEM
// instruction count to keep HBM pipes saturated).
// ---------------------------------------------------------------------------
__global__ void gcn_gemv_kernel(const float* __restrict__ x,
                                const float* __restrict__ w_gcn,
                                const float* __restrict__ b_gcn,
                                float* __restrict__ feat) {
    __shared__ float wsm[ROI];
    const int tid = threadIdx.x;
    for (int j = tid; j < ROI; j += blockDim.x) wsm[j] = w_gcn[j];
    __syncthreads();

    const int blk = blockIdx.x;                       // b*T + t
    const float* base = x + (size_t)blk * (ROI * ROI);
    const float b0 = b_gcn[0];
    const int wave = tid >> 5, lane = tid & 31;

    for (int row = wave; row < ROI; row += 8) {
        const v4f* rp4 = (const v4f*)(base + (size_t)row * ROI);
        __builtin_prefetch(base + (size_t)(row + 8) * ROI, 0, 1);
        float s = 0.0f;
        for (int j4 = lane; j4 < ROI / 4; j4 += 32) {     // 50 chunks of float4
            const v4f v = __builtin_nontemporal_load(rp4 + j4);
            const int j = 4 * j4;
            s += v.x * wsm[j] + v.y * wsm[j + 1] + v.z * wsm[j + 2] + v.w * wsm[j + 3];
        }
        #pragma unroll
        for (int off = 16; off > 0; off >>= 1)
            s += __shfl_down(s, off, 32);
        if (lane == 0) {
            const float v = s + b0;
            feat[(size_t)blk * ROI + row] = v > 0.0f ? v : 0.0f;
        }
    }
}

// ---------------------------------------------------------------------------
// K2: xg[6400,400] = feat[6400,200] @ W_ih^T[200,400] + (b_ih + b_hh)
// One wave per 16x16 output tile, fp32 WMMA 16x16x4, K-loop of 50 steps.
// 10000 tiles total = 2500 blocks x 4 waves.
// ---------------------------------------------------------------------------
__global__ void xg_gemm_kernel(const float* __restrict__ feat,
                               const float* __restrict__ W_ih,   // [400,200]
                               const float* __restrict__ b_ih,
                               const float* __restrict__ b_hh,
                               float* __restrict__ xg) {
    const int wave = threadIdx.x >> 5;
    const int lane = threadIdx.x & 31;
    const int tile = blockIdx.x * 4 + wave;           // 0..9999
    const int mT = tile / 25, nT = tile % 25;

    const int mn   = lane & 15;
    const int koff = (lane >> 4) * 2;                 // 0 or 2
    const int row  = mT * 16 + mn;                    // A row for this lane
    const int col  = nT * 16 + mn;                    // B col for this lane

    v8f acc = {};
    for (int k = 0; k < ROI; k += 4) {
        v2f a, b;
        // A: 16x4 f32 fragment (lanes 0-15 hold K={0,1}, lanes 16-31 K={2,3})
        a.x = feat[(size_t)row * ROI + k + koff];
        a.y = feat[(size_t)row * ROI + k + koff + 1];
        // B: 4x16 fragment; B[k][g] = W_ih[g][k]
        b.x = W_ih[(size_t)col * ROI + k + koff];
        b.y = W_ih[(size_t)col * ROI + k + koff + 1];
        acc = __builtin_amdgcn_wmma_f32_16x16x4_f32(
            false, a, false, b, (short)0, acc, false, false);
    }
    const float cb = b_ih[col] + b_hh[col];
    const int mbase = mT * 16 + (lane >> 4) * 8;      // D: VGPR r -> M = r (+8)
    #pragma unroll
    for (int r = 0; r < 8; ++r)
        xg[(size_t)(mbase + r) * G4 + col] = acc[r] + cb;
}

// ---------------------------------------------------------------------------
// K3: LSTM recurrence. Single persistent workgroup (32 waves).
// LDS residency (dynamic): h (25.6K) + c (25.6K) + gates (102.4K) +
// W_hh (160K) = 313.6 KB < 320 KB/WGP.
// W_hh is DMA'd into LDS ONCE via global_load_async_to_lds_b128 (ASYNCcnt),
// so the 100 sequential steps read B-fragments from LDS only.
// Per step: gates = xg[:,t,:] + h @ W_hh^T via 100 fp32 WMMA 16x16 tiles,
// then fused sigmoid/tanh cell update; hs written to global.
// ---------------------------------------------------------------------------
__global__ void lstm_kernel(const float* __restrict__ xg,       // [6400,400], row = b*T+t
                            const float* __restrict__ W_hh,     // [400,100]
                            float* __restrict__ hs) {           // [B,T,H]
    extern __shared__ float smem[];
    float* h_lds = smem;                            // B*H   = 6400
    float* c_lds = smem + BB * HH;                  // B*H   = 6400
    float* g_lds = smem + 2 * BB * HH;              // B*G4  = 25600
    float* w_lds = smem + 2 * BB * HH + BB * G4;    // 400*100 = 40000

    const int tid  = threadIdx.x;
    const int wave = tid >> 5, lane = tid & 31;

    // ---- one-time async DMA of W_hh (160 KB) into LDS ----
    // LDS byte address comes from the VDST VGPR; dynamic LDS starts at 0
    // (no static LDS in this kernel), so the byte offset of w_lds is exact.
    {
        const unsigned wbase = (unsigned)((2 * BB * HH + BB * G4) * sizeof(float));
        for (int i = tid; i < (G4 * HH) / 4; i += blockDim.x) {   // 10000 x float4
            const unsigned dst = wbase + (unsigned)i * 16u;
            const float* src = W_hh + (size_t)i * 4;
            asm volatile("global_load_async_to_lds_b128 %0, %1, off"
                         :: "v"(dst), "v"(src) : "memory");
        }
        asm volatile("s_wait_asynccnt 0x0" ::: "memory");
    }
    for (int i = tid; i < BB * HH; i += blockDim.x) { h_lds[i] = 0.0f; c_lds[i] = 0.0f; }
    __syncthreads();

    for (int t = 0; t < TT; ++t) {
        // --- gates = xg[:,t,:] + h @ W_hh^T  (M=64, N=400, K=100) ---
        for (int tile = wave; tile < 100; tile += 32) {   // 4 x 25 tiles
            const int mT = tile / 25, nT = tile % 25;
            const int mn   = lane & 15;
            const int koff = (lane >> 4) * 2;
            const int brow = mT * 16 + mn;
            const int col  = nT * 16 + mn;
            v8f acc = {};
            for (int k = 0; k < HH; k += 4) {
                v2f a, b;
                a.x = h_lds[brow * HH + k + koff];
                a.y = h_lds[brow * HH + k + koff + 1];
                b.x = w_lds[col * HH + k + koff];
                b.y = w_lds[col * HH + k + koff + 1];
                acc = __builtin_amdgcn_wmma_f32_16x16x4_f32(
                    false, a, false, b, (short)0, acc, false, false);
            }
            const int mbase = mT * 16 + (lane >> 4) * 8;
            #pragma unroll
            for (int r = 0; r < 8; ++r) {
                const int bidx = mbase + r;
                g_lds[bidx * G4 + col] =
                    acc[r] + xg[((size_t)bidx * TT + t) * G4 + col];
            }
        }
        __syncthreads();

        // --- elementwise cell update (torch gate order i,f,g,o) ---
        for (int idx = tid; idx < BB * HH; idx += blockDim.x) {
            const int b = idx / HH, hh = idx % HH;
            const float gi = g_lds[b * G4 + hh];
            const float gf = g_lds[b * G4 + HH + hh];
            const float gg = g_lds[b * G4 + 2 * HH + hh];
            const float go = g_lds[b * G4 + 3 * HH + hh];
            const float cc = sigmoidf_(gf) * c_lds[idx] + sigmoidf_(gi) * tanhf(gg);
            const float hv = sigmoidf_(go) * tanhf(cc);
            c_lds[idx] = cc;
            h_lds[idx] = hv;
            hs[(size_t)b * TT * HH + (size_t)t * HH + hh] = hv;
        }
        __syncthreads();
    }
}

// ---------------------------------------------------------------------------
// K4a: batch-norm stats over the raw-reshaped [6400, T] view of hs.
// (reshape is trivial: X[r][c] == hs_flat[r*100 + c])
// ---------------------------------------------------------------------------
__global__ void bng_stats_kernel(const float* __restrict__ hs,
                                 float* __restrict__ mean,
                                 float* __restrict__ rstd) {
    __shared__ float s1[256], s2[256];
    const int c = blockIdx.x;          // column 0..T-1
    const int tid = threadIdx.x;
    float a = 0.0f, q = 0.0f;
    for (int r = tid; r < NROWS; r += 256) {
        const float v = hs[(size_t)r * TT + c];
        a += v; q += v * v;
    }
    s1[tid] = a; s2[tid] = q;
    __syncthreads();
    for (int off = 128; off > 0; off >>= 1) {
        if (tid < off) { s1[tid] += s1[tid + off]; s2[tid] += s2[tid + off]; }
        __syncthreads();
    }
    if (tid == 0) {
        const float m = s1[0] / (float)NROWS;
        const float v = s2[0] / (float)NROWS - m * m;   // biased var (torch train)
        mean[c] = m;
        rstd[c] = rsqrtf(v + EPSV);
    }
}

// K4b: out1[r] = sum_c BN(X[r,c]) * fl1_w[c] + fl1_b   (also == mat2[b,h])
__global__ void fl1_kernel(const float* __restrict__ hs,
                           const float* __restrict__ mean,
                           const float* __restrict__ rstd,
                           const float* __restrict__ bng_g,
                           const float* __restrict__ bng_b,
                           const float* __restrict__ fl1_w,
                           const float* __restrict__ fl1_b,
                           float* __restrict__ out1) {
    const int r = blockIdx.x * blockDim.x + threadIdx.x;
    if (r >= NROWS) return;
    float acc = 0.0f;
    for (int c = 0; c < TT; ++c) {
        const float v = (hs[(size_t)r * TT + c] - mean[c]) * rstd[c] * bng_g[c] + bng_b[c];
        acc += v * fl1_w[c];
    }
    out1[r] = acc + fl1_b[0];
}

// K4c: BN+relu -> fc2 -> BN+relu -> fc3 -> softmax. Tiny; single block.
__global__ void head_kernel(const float* __restrict__ out1,     // [B,H] (=[64,100])
                            const float* __restrict__ bn1_g, const float* __restrict__ bn1_b,
                            const float* __restrict__ fl2_w, const float* __restrict__ fl2_b,
                            const float* __restrict__ bn2_g, const float* __restrict__ bn2_b,
                            const float* __restrict__ fl3_w, const float* __restrict__ fl3_b,
                            float* __restrict__ out) {
    __shared__ float z1[BB * HH];     // 6400
    __shared__ float z2[BB * 64];     // 4096
    __shared__ float m1[HH], r1[HH], m2[64], r2[64];
    const int tid = threadIdx.x;      // 128 threads

    for (int i = tid; i < BB * HH; i += 128) z1[i] = out1[i];
    __syncthreads();

    for (int f = tid; f < HH; f += 128) {
        float a = 0.0f, q = 0.0f;
        for (int b = 0; b < BB; ++b) { const float v = z1[b * HH + f]; a += v; q += v * v; }
        const float m = a / (float)BB, v = q / (float)BB - m * m;
        m1[f] = m; r1[f] = rsqrtf(v + EPSV);
    }
    __syncthreads();
    for (int i = tid; i < BB * HH; i += 128) {
        const int f = i % HH;
        const float v = (z1[i] - m1[f]) * r1[f] * bn1_g[f] + bn1_b[f];
        z1[i] = v > 0.0f ? v : 0.0f;
    }
    __syncthreads();

    for (int i = tid; i < BB * 64; i += 128) {
        const int b = i / 64, j = i % 64;
        float a = fl2_b[j];
        for (int k = 0; k < HH; ++k) a += z1[b * HH + k] * fl2_w[j * HH + k];
        z2[i] = a;
    }
    __syncthreads();
    for (int j = tid; j < 64; j += 128) {
        float a = 0.0f, q = 0.0f;
        for (int b = 0; b < BB; ++b) { const float v = z2[b * 64 + j]; a += v; q += v * v; }
        const float m = a / (float)BB, v = q / (float)BB - m * m;
        m2[j] = m; r2[j] = rsqrtf(v + EPSV);
    }
    __syncthreads();
    for (int i = tid; i < BB * 64; i += 128) {
        const int j = i % 64;
        const float v = (z2[i] - m2[j]) * r2[j] * bn2_g[j] + bn2_b[j];
        z2[i] = v > 0.0f ? v : 0.0f;
    }
    __syncthreads();

    for (int b = tid; b < BB; b += 128) {
        float l0 = fl3_b[0], l1 = fl3_b[1];
        for (int j = 0; j < 64; ++j) {
            l0 += z2[b * 64 + j] * fl3_w[j];
            l1 += z2[b * 64 + j] * fl3_w[64 + j];
        }
        const float mx = fmaxf(l0, l1);
        const float e0 = __expf(l0 - mx), e1 = __expf(l1 - mx);
        const float s = e0 + e1;
        out[b * 2 + 0] = e0 / s;
        out[b * 2 + 1] = e1 / s;
    }
}

// ---------------------------------------------------------------------------
extern "C" void kernel_launch(void* const* d_in, const int* in_sizes, int n_in,
                              void* d_out, int out_size, void* d_ws, size_t ws_size,
                              hipStream_t stream) {
    const float* x      = (const float*)d_in[0];
    const float* w_gcn  = (const float*)d_in[1];
    const float* b_gcn  = (const float*)d_in[2];
    const float* W_ih   = (const float*)d_in[3];
    const float* W_hh   = (const float*)d_in[4];
    const float* b_ih   = (const float*)d_in[5];
    const float* b_hh   = (const float*)d_in[6];
    const float* bng_g  = (const float*)d_in[7];
    const float* bng_b  = (const float*)d_in[8];
    const float* fl1_w  = (const float*)d_in[9];
    const float* fl1_b  = (const float*)d_in[10];
    const float* bn1_g  = (const float*)d_in[11];
    const float* bn1_b  = (const float*)d_in[12];
    const float* fl2_w  = (const float*)d_in[13];
    const float* fl2_b  = (const float*)d_in[14];
    const float* bn2_g  = (const float*)d_in[15];
    const float* bn2_b  = (const float*)d_in[16];
    const float* fl3_w  = (const float*)d_in[17];
    const float* fl3_b  = (const float*)d_in[18];

    float* ws   = (float*)d_ws;
    float* feat = ws;                                   // 6400*200
    float* xg   = feat + (size_t)NROWS * ROI;           // 6400*400
    float* hs   = xg + (size_t)NROWS * G4;              // 64*100*100
    float* mean = hs + (size_t)BB * TT * HH;            // T
    float* rstd = mean + TT;                            // T
    float* out1 = rstd + TT;                            // 6400

    // K1: stream x (1.024 GB) once — bandwidth bound (~44us at 23.3 TB/s)
    gcn_gemv_kernel<<<NROWS, 256, 0, stream>>>(x, w_gcn, b_gcn, feat);

    // K2: input-gate GEMM via fp32 WMMA (10000 tiles, 4 waves/block)
    xg_gemm_kernel<<<2500, 128, 0, stream>>>(feat, W_ih, b_ih, b_hh, xg);

    // K3: persistent single-WGP LSTM; dynamic LDS = h + c + gates + W_hh
    const size_t lstm_lds =
        (size_t)(2 * BB * HH + BB * G4 + G4 * HH) * sizeof(float); // 313.6 KB
    lstm_kernel<<<1, 1024, lstm_lds, stream>>>(xg, W_hh, hs);

    // K4: head
    bng_stats_kernel<<<TT, 256, 0, stream>>>(hs, mean, rstd);
    fl1_kernel<<<(NROWS + 255) / 256, 256, 0, stream>>>(hs, mean, rstd, bng_g, bng_b,
                                                        fl1_w, fl1_b, out1);
    head_kernel<<<1, 128, 0, stream>>>(out1, bn1_g, bn1_b, fl2_w, fl2_b,
                                       bn2_g, bn2_b, fl3_w, fl3_b, (float*)d_out);
}